// GATMultiClass_48086453846345
// MI455X (gfx1250) — compile-verified
//
#include <hip/hip_runtime.h>
#include <hip/hip_bf16.h>

typedef __attribute__((ext_vector_type(2))) float v2f;
typedef __attribute__((ext_vector_type(8))) float v8f;

namespace {

constexpr int F_IN  = 128;
constexpr int HEADS = 4;
constexpr int CCH   = 64;
constexpr int HC    = HEADS * CCH;   // 256
constexpr int UDIM  = 32;
constexpr int NCOUT = 6;
constexpr float EPS = 1e-3f;

__device__ __forceinline__ void atomic_add_f32(float* p, float v) {
  __hip_atomic_fetch_add(p, v, __ATOMIC_RELAXED, __HIP_MEMORY_SCOPE_AGENT);
}

// float max via int/uint atomics (valid for mixed signs when init = -inf)
__device__ __forceinline__ void atomic_max_f32(float* p, float v) {
  if (v >= 0.0f)
    __hip_atomic_fetch_max(reinterpret_cast<int*>(p), __float_as_int(v),
                           __ATOMIC_RELAXED, __HIP_MEMORY_SCOPE_AGENT);
  else
    __hip_atomic_fetch_min(reinterpret_cast<unsigned int*>(p), __float_as_uint(v),
                           __ATOMIC_RELAXED, __HIP_MEMORY_SCOPE_AGENT);
}

__global__ __launch_bounds__(256)
void fill_f32(float* __restrict__ p, float v, long long n) {
  long long t = (long long)blockIdx.x * blockDim.x + threadIdx.x;
  if (t < n) p[t] = v;
}

// out[M x 256] = A[M x K] @ W[K x 256], fp32 via V_WMMA_F32_16X16X4_F32.
// One wave per 16x16 output tile. Documented layouts:
//   A 16x4: lane = m + 16*(k/2), vgpr = k%2
//   B 4x16: lane = n + 16*(k/2), vgpr = k%2
//   C/D 16x16: vgpr i, lanes 0-15 -> M=i, lanes 16-31 -> M=i+8
__global__ __launch_bounds__(256)
void gemm_wmma_f32(const float* __restrict__ A, const float* __restrict__ W,
                   float* __restrict__ out, int M, int K) {
  int wave = blockIdx.x * (blockDim.x >> 5) + (threadIdx.x >> 5);
  int lane = threadIdx.x & 31;
  int mt = wave >> 4;            // 16 N-tiles (256/16) per M-tile
  int nt = wave & 15;
  if (mt * 16 >= M) return;      // wave-uniform
  int half = lane >> 4;
  int r    = lane & 15;
  int m0 = mt * 16, n0 = nt * 16;
  int mrow = m0 + r; if (mrow > M - 1) mrow = M - 1;   // clamp (M%16==0 here anyway)
  const float* arow = A + (long long)mrow * K;
  const float* wcol = W + n0 + r;
  v8f acc = {};
  for (int k0 = 0; k0 < K; k0 += 4) {
    int ka = k0 + 2 * half;
    v2f a, b;
    a.x = arow[ka];
    a.y = arow[ka + 1];
    b.x = wcol[(long long)ka * HC];
    b.y = wcol[(long long)(ka + 1) * HC];
    acc = __builtin_amdgcn_wmma_f32_16x16x4_f32(false, a, false, b,
                                                (short)0, acc, false, false);
  }
  long long obase = (long long)(m0 + 8 * half) * HC + n0 + r;
#pragma unroll
  for (int i = 0; i < 8; ++i) {
    int row = m0 + 8 * half + i;
    if (row < M) out[obase + (long long)i * HC] = acc[i];
  }
}

// e_self[n,h] = <h[n,h,:], a_self[h,:]>, e_neigh likewise. t = n*HEADS+head.
__global__ __launch_bounds__(256)
void attn_logits(const float* __restrict__ h, const float* __restrict__ a_self,
                 const float* __restrict__ a_neigh,
                 float* __restrict__ e_self, float* __restrict__ e_neigh, int N) {
  long long t = (long long)blockIdx.x * blockDim.x + threadIdx.x;
  if (t >= (long long)N * HEADS) return;
  int head = (int)(t & (HEADS - 1));
  const float4* hv = reinterpret_cast<const float4*>(h + t * CCH);
  const float4* as = reinterpret_cast<const float4*>(a_self + head * CCH);
  const float4* an = reinterpret_cast<const float4*>(a_neigh + head * CCH);
  float ss = 0.f, sn = 0.f;
#pragma unroll
  for (int i = 0; i < CCH / 4; ++i) {
    float4 x = hv[i], s = as[i], n = an[i];
    ss += x.x * s.x + x.y * s.y + x.z * s.z + x.w * s.w;
    sn += x.x * n.x + x.y * n.y + x.z * n.z + x.w * n.w;
  }
  e_self[t]  = ss;
  e_neigh[t] = sn;
}

__device__ __forceinline__ void edge_nodes(int e, int E, const int* src,
                                           const int* dst, int& s, int& d) {
  if (e < E) { s = src[e]; d = dst[e]; } else { s = e - E; d = s; }  // self-loops
}

__global__ __launch_bounds__(256)
void edge_max(const int* __restrict__ src, const int* __restrict__ dst,
              const float* __restrict__ e_self, const float* __restrict__ e_neigh,
              float* __restrict__ emax, int E, int N) {
  long long t = (long long)blockIdx.x * blockDim.x + threadIdx.x;
  if (t >= (long long)(E + N) * HEADS) return;
  int head = (int)(t & (HEADS - 1));
  int e = (int)(t >> 2);
  int s, d; edge_nodes(e, E, src, dst, s, d);
  float v = e_self[(long long)d * HEADS + head] + e_neigh[(long long)s * HEADS + head];
  v = v > 0.f ? v : 0.2f * v;   // leaky_relu(., 0.2)
  atomic_max_f32(&emax[(long long)d * HEADS + head], v);
}

__global__ __launch_bounds__(256)
void edge_expsum(const int* __restrict__ src, const int* __restrict__ dst,
                 const float* __restrict__ e_self, const float* __restrict__ e_neigh,
                 const float* __restrict__ emax, float* __restrict__ denom,
                 float* __restrict__ alpha, int E, int N) {
  long long t = (long long)blockIdx.x * blockDim.x + threadIdx.x;
  if (t >= (long long)(E + N) * HEADS) return;
  int head = (int)(t & (HEADS - 1));
  int e = (int)(t >> 2);
  int s, d; edge_nodes(e, E, src, dst, s, d);
  float v = e_self[(long long)d * HEADS + head] + e_neigh[(long long)s * HEADS + head];
  v = v > 0.f ? v : 0.2f * v;
  float ee = expf(v - emax[(long long)d * HEADS + head]);
  alpha[t] = ee;                          // un-normalized weight, reused in pass 3
  atomic_add_f32(&denom[(long long)d * HEADS + head], ee);
}

// 16 lanes per (edge, head): lane quad handles 4 channels (float4 gather + 4 scatter-adds)
__global__ __launch_bounds__(256)
void edge_aggregate(const int* __restrict__ src, const int* __restrict__ dst,
                    const float* __restrict__ alpha, const float* __restrict__ denom,
                    const float* __restrict__ h, float* __restrict__ agg, int E, int N) {
  long long t = (long long)blockIdx.x * blockDim.x + threadIdx.x;
  if (t >= (long long)(E + N) * HEADS * 16) return;
  int quad = (int)(t & 15);
  int eh   = (int)(t >> 4);
  int head = eh & (HEADS - 1);
  int e    = eh >> 2;
  int s, d; edge_nodes(e, E, src, dst, s, d);
  float a = alpha[eh] / denom[(long long)d * HEADS + head];
  float4 hv = *reinterpret_cast<const float4*>(
      h + ((long long)s * HEADS + head) * CCH + quad * 4);
  float* ap = agg + ((long long)d * HEADS + head) * CCH + quad * 4;
  atomic_add_f32(ap + 0, a * hv.x);
  atomic_add_f32(ap + 1, a * hv.y);
  atomic_add_f32(ap + 2, a * hv.z);
  atomic_add_f32(ap + 3, a * hv.w);
}

// mean over heads + bias -> relu -> batchnorm (inference)
__global__ __launch_bounds__(256)
void finalize_node(const float* __restrict__ agg, const float* __restrict__ bias,
                   const float* __restrict__ gamma, const float* __restrict__ beta,
                   const float* __restrict__ mean, const float* __restrict__ var,
                   float* __restrict__ out, int N) {
  long long t = (long long)blockIdx.x * blockDim.x + threadIdx.x;
  if (t >= (long long)N * CCH) return;
  int c = (int)(t & (CCH - 1));
  long long n = t >> 6;
  const float* ap = agg + n * HC;
  float v = 0.25f * (ap[c] + ap[CCH + c] + ap[2 * CCH + c] + ap[3 * CCH + c]) + bias[c];
  v = fmaxf(v, 0.f);
  v = (v - mean[c]) * rsqrtf(var[c] + EPS) * gamma[c] + beta[c];
  out[t] = v;
}

// seg is sorted: run-length accumulate locally, one atomic per run boundary
__global__ __launch_bounds__(256)
void pool_seg(const float* __restrict__ x, const int* __restrict__ seg,
              float* __restrict__ pooled, int N) {
  const int CHUNK = 128;
  int t = blockIdx.x * blockDim.x + threadIdx.x;
  int c = t & (CCH - 1);
  int chunk = t >> 6;
  long long n0 = (long long)chunk * CHUNK;
  if (n0 >= N) return;
  long long n1 = n0 + CHUNK; if (n1 > N) n1 = N;
  int cur = seg[n0];
  float acc = 0.f;
  for (long long n = n0; n < n1; ++n) {
    int sg = seg[n];
    if (sg != cur) {
      atomic_add_f32(&pooled[(long long)cur * CCH + c], acc);
      acc = 0.f; cur = sg;
    }
    acc += x[n * CCH + c];
  }
  atomic_add_f32(&pooled[(long long)cur * CCH + c], acc);
}

__global__ __launch_bounds__(256)
void dense1(const float* __restrict__ pooled, const float* __restrict__ u,
            const float* __restrict__ w1, const float* __restrict__ b1,
            float* __restrict__ z1, int B) {
  int t = blockIdx.x * blockDim.x + threadIdx.x;
  if (t >= B * 128) return;
  int j = t & 127, b = t >> 7;
  float s = b1[j];
  for (int k = 0; k < CCH; ++k)  s += pooled[b * CCH + k] * w1[k * 128 + j];
  for (int k = 0; k < UDIM; ++k) s += u[b * UDIM + k] * w1[(CCH + k) * 128 + j];
  z1[t] = fmaxf(s, 0.f);
}

__global__ __launch_bounds__(256)
void dense2(const float* __restrict__ z1, const float* __restrict__ w2,
            const float* __restrict__ b2, float* __restrict__ z2, int B) {
  int t = blockIdx.x * blockDim.x + threadIdx.x;
  if (t >= B * CCH) return;
  int j = t & (CCH - 1), b = t >> 6;
  float s = b2[j];
  for (int k = 0; k < 128; ++k) s += z1[b * 128 + k] * w2[k * CCH + j];
  z2[t] = fmaxf(s, 0.f);
}

__global__ __launch_bounds__(64)
void dense_out(const float* __restrict__ z2, const float* __restrict__ wo,
               const float* __restrict__ bo, float* __restrict__ out, int B) {
  int b = blockIdx.x * blockDim.x + threadIdx.x;
  if (b >= B) return;
  float lg[NCOUT];
  float m = -3.4e38f;
  for (int j = 0; j < NCOUT; ++j) {
    float s = bo[j];
    for (int k = 0; k < CCH; ++k) s += z2[b * CCH + k] * wo[k * NCOUT + j];
    lg[j] = s; m = fmaxf(m, s);
  }
  float se = 0.f;
  for (int j = 0; j < NCOUT; ++j) { lg[j] = expf(lg[j] - m); se += lg[j]; }
  float inv = 1.f / se;
  for (int j = 0; j < NCOUT; ++j) out[b * NCOUT + j] = lg[j] * inv;
}

} // namespace

extern "C" void kernel_launch(void* const* d_in, const int* in_sizes, int n_in,
                              void* d_out, int out_size, void* d_ws, size_t ws_size,
                              hipStream_t stream) {
  const float* x    = (const float*)d_in[0];
  const int*   src  = (const int*)d_in[1];
  const int*   dst  = (const int*)d_in[2];
  const int*   seg  = (const int*)d_in[3];
  const float* u    = (const float*)d_in[4];
  const float* g1k  = (const float*)d_in[5];
  const float* g1as = (const float*)d_in[6];
  const float* g1an = (const float*)d_in[7];
  const float* g1b  = (const float*)d_in[8];
  const float* g2k  = (const float*)d_in[9];
  const float* g2as = (const float*)d_in[10];
  const float* g2an = (const float*)d_in[11];
  const float* g2b  = (const float*)d_in[12];
  const float* bn1g = (const float*)d_in[13];
  const float* bn1b = (const float*)d_in[14];
  const float* bn1m = (const float*)d_in[15];
  const float* bn1v = (const float*)d_in[16];
  const float* bn2g = (const float*)d_in[17];
  const float* bn2b = (const float*)d_in[18];
  const float* bn2m = (const float*)d_in[19];
  const float* bn2v = (const float*)d_in[20];
  const float* w1   = (const float*)d_in[21];
  const float* b1   = (const float*)d_in[22];
  const float* w2   = (const float*)d_in[23];
  const float* b2   = (const float*)d_in[24];
  const float* wo   = (const float*)d_in[25];
  const float* bo   = (const float*)d_in[26];
  float* out = (float*)d_out;
  (void)n_in; (void)out_size; (void)ws_size;

  const int N  = in_sizes[0] / F_IN;
  const int E  = in_sizes[1];
  const int B  = in_sizes[4] / UDIM;
  const int E2 = E + N;

  size_t off = 0;
  auto alloc = [&](size_t bytes) -> float* {
    float* p = (float*)((char*)d_ws + off);
    off += (bytes + 255) & ~(size_t)255;
    return p;
  };
  float* h      = alloc((size_t)N * HC * 4);
  float* agg    = alloc((size_t)N * HC * 4);
  float* xnode  = alloc((size_t)N * CCH * 4);
  float* es     = alloc((size_t)N * HEADS * 4);
  float* en     = alloc((size_t)N * HEADS * 4);
  float* emax   = alloc((size_t)N * HEADS * 4);
  float* denom  = alloc((size_t)N * HEADS * 4);
  float* alpha  = alloc((size_t)E2 * HEADS * 4);
  float* pooled = alloc((size_t)B * CCH * 4);
  float* z1     = alloc((size_t)B * 128 * 4);
  float* z2     = alloc((size_t)B * CCH * 4);

  auto cdiv = [](long long a, long long b) { return (int)((a + b - 1) / b); };

  const int mtiles      = (N + 15) / 16;
  const int gemm_blocks = cdiv((long long)mtiles * 16, 8);  // 8 waves/block
  const long long nh    = (long long)N * HEADS;
  const long long eh    = (long long)E2 * HEADS;
  const float NEG_INF   = -__builtin_inff();

  // ---------------- GAT layer 1 ----------------
  gemm_wmma_f32<<<gemm_blocks, 256, 0, stream>>>(x, g1k, h, N, F_IN);
  attn_logits<<<cdiv(nh, 256), 256, 0, stream>>>(h, g1as, g1an, es, en, N);
  fill_f32<<<cdiv(nh, 256), 256, 0, stream>>>(emax, NEG_INF, nh);
  hipMemsetAsync(denom, 0, (size_t)nh * 4, stream);
  hipMemsetAsync(agg, 0, (size_t)N * HC * 4, stream);
  edge_max<<<cdiv(eh, 256), 256, 0, stream>>>(src, dst, es, en, emax, E, N);
  edge_expsum<<<cdiv(eh, 256), 256, 0, stream>>>(src, dst, es, en, emax, denom, alpha, E, N);
  edge_aggregate<<<cdiv(eh * 16, 256), 256, 0, stream>>>(src, dst, alpha, denom, h, agg, E, N);
  finalize_node<<<cdiv((long long)N * CCH, 256), 256, 0, stream>>>(
      agg, g1b, bn1g, bn1b, bn1m, bn1v, xnode, N);

  // ---------------- GAT layer 2 ----------------
  gemm_wmma_f32<<<gemm_blocks, 256, 0, stream>>>(xnode, g2k, h, N, CCH);
  attn_logits<<<cdiv(nh, 256), 256, 0, stream>>>(h, g2as, g2an, es, en, N);
  fill_f32<<<cdiv(nh, 256), 256, 0, stream>>>(emax, NEG_INF, nh);
  hipMemsetAsync(denom, 0, (size_t)nh * 4, stream);
  hipMemsetAsync(agg, 0, (size_t)N * HC * 4, stream);
  edge_max<<<cdiv(eh, 256), 256, 0, stream>>>(src, dst, es, en, emax, E, N);
  edge_expsum<<<cdiv(eh, 256), 256, 0, stream>>>(src, dst, es, en, emax, denom, alpha, E, N);
  edge_aggregate<<<cdiv(eh * 16, 256), 256, 0, stream>>>(src, dst, alpha, denom, h, agg, E, N);
  finalize_node<<<cdiv((long long)N * CCH, 256), 256, 0, stream>>>(
      agg, g2b, bn2g, bn2b, bn2m, bn2v, xnode, N);

  // ---------------- pool + dense head ----------------
  hipMemsetAsync(pooled, 0, (size_t)B * CCH * 4, stream);
  long long pool_threads = (long long)CCH * cdiv(N, 128);
  pool_seg<<<cdiv(pool_threads, 256), 256, 0, stream>>>(xnode, seg, pooled, N);
  dense1<<<cdiv((long long)B * 128, 256), 256, 0, stream>>>(pooled, u, w1, b1, z1, B);
  dense2<<<cdiv((long long)B * CCH, 256), 256, 0, stream>>>(z1, w2, b2, z2, B);
  dense_out<<<cdiv(B, 64), 64, 0, stream>>>(z2, wo, bo, out, B);
}